// CayleyConv2d_39195871544087
// MI455X (gfx1250) — compile-verified
//
#include <hip/hip_runtime.h>
#include <math.h>

// ---------------------------------------------------------------------------
// CayleyConv2d on gfx1250 (MI455X).
//   x:(32,128,64,64) f32, w:(128,128,3,3), bias:(128), alpha scalar.
//   Pipeline: setup DFT mats -> ||wfft|| reduce -> scale -> rfft2 (WMMA GEMM)
//             -> per-bin Cayley solve:
//                  RHS = 2X - M@X          (fp32 WMMA, K=128)
//                  blocked LU, NB=16:      panel scalar, trailing update WMMA
//                  blocked back-sub:       in-block scalar, cross-block WMMA
//             -> irfft2 (WMMA GEMM) + bias.
// Workspace (floats): ~34.6M floats (~139 MB).
// ---------------------------------------------------------------------------

typedef __attribute__((ext_vector_type(2))) float v2f;
typedef __attribute__((ext_vector_type(8))) float v8f;

#define NN 64
#define NH 33
#define NHP 48
#define BINS (NN * NH)      // 2112
#define CH 128
#define BATCH 32
#define CB (CH * BATCH)     // 4096
#define TWO_PI_OVER_N 0.09817477042468103f  // 2*pi/64

// workspace float offsets
#define WS_ERE 0
#define WS_EIM 4096
#define WS_GRE 8192
#define WS_GIM 11264
#define WS_CRE 14336
#define WS_CIM 18432
#define WS_D1 22528
#define WS_D2 25600
#define WS_NORM 28672
#define WS_SCALE 28673
#define WS_XFRE 28688
#define WS_XFIM (WS_XFRE + BINS * CB)   // +8650752
#define WS_YFRE (WS_XFIM + BINS * CB)
#define WS_YFIM (WS_YFRE + BINS * CB)

// ---------------------------------------------------------------------------
// fp32 WMMA 16x16 tile helper: acc += A(16xK, row-major lda) * B(Kx16, ldb).
// Fragment layouts per CDNA5 ISA 7.12.2 (wave32):
//   A 16x4: lane<16 -> {A[m][0],A[m][1]}, lane>=16 -> {A[m][2],A[m][3]}
//   B 4x16: vgpr0 = rows {0,2}, vgpr1 = rows {1,3}, col = lane%16
//   C/D:    vgpr v = row v + 8*(lane/16), col = lane%16
// ---------------------------------------------------------------------------
__device__ __forceinline__ v8f wmma_k(const float* __restrict__ A, int lda,
                                      const float* __restrict__ B, int ldb,
                                      int K, v8f acc) {
  const int lane = threadIdx.x & 31;
  const int half = lane >> 4;
  const int mr = lane & 15;
  const float* arow = A + mr * lda + 2 * half;
  const float* bcol = B + mr + 2 * half * ldb;
  for (int k0 = 0; k0 < K; k0 += 4) {
    v2f a, b;
    a.x = arow[k0];
    a.y = arow[k0 + 1];
    b.x = bcol[k0 * ldb];
    b.y = bcol[k0 * ldb + ldb];
    acc = __builtin_amdgcn_wmma_f32_16x16x4_f32(false, a, false, b,
                                                (short)0, acc, false, false);
  }
  return acc;
}

// ---------------------------------------------------------------------------
// Kernel 0: build DFT matrices + zero the norm accumulator.
// ---------------------------------------------------------------------------
__global__ void k_setup(float* __restrict__ ws) {
  int g = blockIdx.x * blockDim.x + threadIdx.x;
  int stride = gridDim.x * blockDim.x;
  for (int e = g; e < 4096; e += stride) {
    int k = e >> 6, r = e & 63;
    float th = TWO_PI_OVER_N * (float)((k * r) & 63);
    float s, c;
    __sincosf(th, &s, &c);
    ws[WS_ERE + e] = c;
    ws[WS_EIM + e] = -s;
    ws[WS_CRE + e] = c * (1.0f / 64.0f);
    ws[WS_CIM + e] = s * (1.0f / 64.0f);
  }
  for (int e = g; e < 64 * NHP; e += stride) {
    int r = e / NHP, l = e % NHP;
    if (l < NH) {
      float th = TWO_PI_OVER_N * (float)((r * l) & 63);
      float s, c;
      __sincosf(th, &s, &c);
      ws[WS_GRE + e] = c;
      ws[WS_GIM + e] = -s;
    } else {
      ws[WS_GRE + e] = 0.0f;
      ws[WS_GIM + e] = 0.0f;
    }
    int ld = e / 64, p = e % 64;   // D is 48x64 row-major
    if (ld < NH) {
      float a = (ld == 0 || ld == 32) ? (1.0f / 64.0f) : (2.0f / 64.0f);
      float th2 = TWO_PI_OVER_N * (float)((ld * p) & 63);
      float s2, c2;
      __sincosf(th2, &s2, &c2);
      ws[WS_D1 + e] = a * c2;
      ws[WS_D2 + e] = a * s2;
    } else {
      ws[WS_D1 + e] = 0.0f;
      ws[WS_D2 + e] = 0.0f;
    }
  }
  if (g == 0) ws[WS_NORM] = 0.0f;
}

// ---------------------------------------------------------------------------
// Kernel 1: ||wfft||_F^2 (phases drop out of |.|). One block per bin.
// ---------------------------------------------------------------------------
__global__ __launch_bounds__(256) void k_wnorm(const float* __restrict__ w,
                                               float* __restrict__ norm2) {
  int bin = blockIdx.x;
  int k = bin / NH, l = bin % NH;
  float pre[9], pim[9];
#pragma unroll
  for (int u = 0; u < 3; u++)
#pragma unroll
    for (int v = 0; v < 3; v++) {
      float th = TWO_PI_OVER_N * (float)((k * u + l * v) & 63);
      float s, c;
      __sincosf(th, &s, &c);
      pre[u * 3 + v] = c;
      pim[u * 3 + v] = -s;
    }
  float acc = 0.0f;
  for (int pp = threadIdx.x; pp < CH * CH; pp += 256) {
    const float* wp = w + (size_t)pp * 9;
    float fr = 0.0f, fi = 0.0f;
#pragma unroll
    for (int t = 0; t < 9; t++) {
      fr += wp[t] * pre[t];
      fi += wp[t] * pim[t];
    }
    acc += fr * fr + fi * fi;
  }
  __shared__ float red[256];
  red[threadIdx.x] = acc;
  __syncthreads();
  for (int s = 128; s > 0; s >>= 1) {
    if (threadIdx.x < (unsigned)s) red[threadIdx.x] += red[threadIdx.x + s];
    __syncthreads();
  }
  if (threadIdx.x == 0) atomicAdd(norm2, red[0]);
}

__global__ void k_scale(const float* __restrict__ alpha, float* __restrict__ ws) {
  float n2 = ws[WS_NORM];
  ws[WS_SCALE] = alpha[0] * rsqrtf(fmaxf(n2, 1e-30f));
}

// ---------------------------------------------------------------------------
// Kernel 2: rfft2 of x via WMMA GEMMs. One WG per (b,c) image.
// ---------------------------------------------------------------------------
__global__ __launch_bounds__(256) void k_rfft2(
    const float* __restrict__ x, const float* __restrict__ Ere,
    const float* __restrict__ Eim, const float* __restrict__ Gre,
    const float* __restrict__ Gim, float* __restrict__ XFre,
    float* __restrict__ XFim) {
  __shared__ float sX[64 * 64];
  __shared__ float sTre[64 * 64];
  __shared__ float sTim[64 * 64];
  int img = blockIdx.x;            // b*128 + c
  int c0 = img & 127, b0 = img >> 7;
  const float* xp = x + (size_t)img * 4096;
  for (int i = threadIdx.x; i < 4096; i += 256) sX[i] = xp[i];
  __syncthreads();
  int wid = threadIdx.x >> 5, lane = threadIdx.x & 31;
  int half = lane >> 4, mr = lane & 15;
  for (int job = wid; job < 32; job += 8) {
    int plane = job >> 4, t = job & 15, tr = t >> 2, tc = t & 3;
    v8f acc = {};
    acc = wmma_k((plane ? Eim : Ere) + tr * 16 * 64, 64, sX + tc * 16, 64, 64, acc);
    float* D = (plane ? sTim : sTre) + tr * 16 * 64 + tc * 16;
#pragma unroll
    for (int v = 0; v < 8; v++) D[(v + 8 * half) * 64 + mr] = acc[v];
  }
  __syncthreads();
  for (int job = wid; job < 24; job += 8) {
    int plane = job / 12, t = job % 12, tr = t / 3, tc = t % 3;
    v8f p = {};
    if (plane == 0) {  // re = Tre*Gre - Tim*Gim
      v8f q = {};
      p = wmma_k(sTre + tr * 16 * 64, 64, Gre + tc * 16, NHP, 64, p);
      q = wmma_k(sTim + tr * 16 * 64, 64, Gim + tc * 16, NHP, 64, q);
      p = p - q;
    } else {           // im = Tre*Gim + Tim*Gre
      p = wmma_k(sTre + tr * 16 * 64, 64, Gim + tc * 16, NHP, 64, p);
      p = wmma_k(sTim + tr * 16 * 64, 64, Gre + tc * 16, NHP, 64, p);
    }
    float* OUT = plane ? XFim : XFre;
#pragma unroll
    for (int v = 0; v < 8; v++) {
      int kk = tr * 16 + v + 8 * half;
      int l = tc * 16 + mr;
      if (l < NH) OUT[(size_t)(kk * NH + l) * CB + c0 * BATCH + b0] = p[v];
    }
  }
}

// ---------------------------------------------------------------------------
// Kernel 3: per-bin Cayley solve, 160 KB LDS augmented system [I+A | 2X-MX],
// ld=160 (M 128x128 complex, RHS 128x32 complex).
// Blocked LU NB=16: panel scalar, rank-16 trailing updates on fp32 WMMA.
// No pivoting needed: I+A (A skew-Hermitian) has |pivot| >= 1.
// ---------------------------------------------------------------------------
__global__ __launch_bounds__(256) void k_solve(
    const float* __restrict__ w, const float* __restrict__ XFre,
    const float* __restrict__ XFim, const float* __restrict__ scale_p,
    float* __restrict__ YFre, float* __restrict__ YFim) {
  extern __shared__ float smem[];
  float* Are = smem;                 // 128*160
  float* Aim = smem + 128 * 160;     // 128*160
  const int bin = blockIdx.x;
  const int k = bin / NH, l = bin % NH;
  const float s = scale_p[0];
  float pre[9], pim[9];
#pragma unroll
  for (int u = 0; u < 3; u++)
#pragma unroll
    for (int v = 0; v < 3; v++) {
      float th = TWO_PI_OVER_N * (float)((k * u + l * v) & 63);
      float sn, cs;
      __sincosf(th, &sn, &cs);
      pre[u * 3 + v] = cs;
      pim[u * 3 + v] = -sn;
    }
  float cP, sP;
  __sincosf(TWO_PI_OVER_N * (float)((k + l) & 63), &sP, &cP);  // P=(cP,-sP)
  // Phase 0: M = I + Wn - Wn^H, Wn = s * P * conj(fftw)
  for (int e = threadIdx.x; e < CH * CH; e += 256) {
    int i = e >> 7, j = e & 127;
    const float* wij = w + (size_t)(i * CH + j) * 9;
    const float* wji = w + (size_t)(j * CH + i) * 9;
    float fr1 = 0, fi1 = 0, fr2 = 0, fi2 = 0;
#pragma unroll
    for (int t = 0; t < 9; t++) {
      float a = wij[t], b = wji[t];
      fr1 += a * pre[t]; fi1 += a * pim[t];
      fr2 += b * pre[t]; fi2 += b * pim[t];
    }
    float wr1 = s * (cP * fr1 - sP * fi1), wi1 = s * (-cP * fi1 - sP * fr1);
    float wr2 = s * (cP * fr2 - sP * fi2), wi2 = s * (-cP * fi2 - sP * fr2);
    Are[i * 160 + j] = wr1 - wr2 + (i == j ? 1.0f : 0.0f);
    Aim[i * 160 + j] = wi1 + wi2;
  }
  // Phase 1: load X into augmented cols 128..159
  for (int e = threadIdx.x; e < CH * BATCH; e += 256) {
    int i = e >> 5, jj = e & 31;
    Are[i * 160 + 128 + jj] = XFre[(size_t)bin * CB + e];
    Aim[i * 160 + 128 + jj] = XFim[(size_t)bin * CB + e];
  }
  __syncthreads();
  // Phase 2: P = M @ X via WMMA (8x2 complex tiles, 2 per wave), K=128
  int wid = threadIdx.x >> 5, lane = threadIdx.x & 31;
  int half = lane >> 4, mr = lane & 15;
  v8f prA[2], piA[2];
#pragma unroll
  for (int jj = 0; jj < 2; jj++) {
    int job = wid + jj * 8;
    int tr = job >> 1, tc = job & 1;
    const float* Ar = Are + tr * 16 * 160;
    const float* Ai = Aim + tr * 16 * 160;
    const float* Xr = Are + 128 + tc * 16;
    const float* Xi = Aim + 128 + tc * 16;
    v8f a1 = {}, a2 = {}, b1 = {}, b2 = {};
    a1 = wmma_k(Ar, 160, Xr, 160, 128, a1);
    a2 = wmma_k(Ai, 160, Xi, 160, 128, a2);
    b1 = wmma_k(Ar, 160, Xi, 160, 128, b1);
    b2 = wmma_k(Ai, 160, Xr, 160, 128, b2);
    prA[jj] = a1 - a2;
    piA[jj] = b1 + b2;
  }
  __syncthreads();
  // Phase 3: RHS = 2X - P
#pragma unroll
  for (int jj = 0; jj < 2; jj++) {
    int job = wid + jj * 8;
    int tr = job >> 1, tc = job & 1;
#pragma unroll
    for (int v = 0; v < 8; v++) {
      int r = tr * 16 + v + 8 * half;
      int cidx = 128 + tc * 16 + mr;
      float xr = Are[r * 160 + cidx], xi = Aim[r * 160 + cidx];
      Are[r * 160 + cidx] = 2.0f * xr - prA[jj][v];
      Aim[r * 160 + cidx] = 2.0f * xi - piA[jj][v];
    }
  }
  __syncthreads();
  // Phase 4: blocked forward elimination, NB=16.
  for (int kb = 0; kb < 8; kb++) {
    const int bs = kb * 16, be = bs + 16;
    // 4a: panel + pivot-row formation (scalar). Per column kk:
    //   scale row kk full width; eliminate in-block rows full width;
    //   eliminate below-block rows over panel columns only (multipliers).
    for (int kk = bs; kk < be; kk++) {
      float prv = Are[kk * 160 + kk], piv = Aim[kk * 160 + kk];
      float inv = 1.0f / (prv * prv + piv * piv);
      float ir = prv * inv, ii = -piv * inv;
      for (int j = kk + 1 + threadIdx.x; j < 160; j += 256) {
        float re = Are[kk * 160 + j], im = Aim[kk * 160 + j];
        Are[kk * 160 + j] = re * ir - im * ii;
        Aim[kk * 160 + j] = re * ii + im * ir;
      }
      __syncthreads();
      for (int r = kk + 1 + wid; r < be; r += 8) {
        float lr = Are[r * 160 + kk], li = Aim[r * 160 + kk];
        for (int j = kk + 1 + lane; j < 160; j += 32) {
          float ur = Are[kk * 160 + j], ui = Aim[kk * 160 + j];
          Are[r * 160 + j] -= lr * ur - li * ui;
          Aim[r * 160 + j] -= lr * ui + li * ur;
        }
      }
      for (int r = be + wid; r < 128; r += 8) {
        float lr = Are[r * 160 + kk], li = Aim[r * 160 + kk];
        int j = kk + 1 + lane;
        if (j < be) {
          float ur = Are[kk * 160 + j], ui = Aim[kk * 160 + j];
          Are[r * 160 + j] -= lr * ur - li * ui;
          Aim[r * 160 + j] -= lr * ui + li * ur;
        }
      }
      __syncthreads();
    }
    // 4b: trailing update A[be:,be:] -= L21 @ U12 (complex, K=16) on WMMA.
    //     Reads (panel cols / pivot rows) disjoint from writes.
    if (be < 128) {
      int nrt = (128 - be) >> 4, nct = (160 - be) >> 4;
      for (int job = wid; job < nrt * nct; job += 8) {
        int tr = job / nct, tc = job % nct;
        const float* Lr = Are + (be + tr * 16) * 160 + bs;
        const float* Li = Aim + (be + tr * 16) * 160 + bs;
        const float* Ur = Are + bs * 160 + be + tc * 16;
        const float* Ui = Aim + bs * 160 + be + tc * 16;
        v8f p1 = {}, p2 = {}, q1 = {}, q2 = {};
        p1 = wmma_k(Lr, 160, Ur, 160, 16, p1);
        p2 = wmma_k(Li, 160, Ui, 160, 16, p2);
        q1 = wmma_k(Lr, 160, Ui, 160, 16, q1);
        q2 = wmma_k(Li, 160, Ur, 160, 16, q2);
#pragma unroll
        for (int v = 0; v < 8; v++) {
          int base = (be + tr * 16 + v + 8 * half) * 160 + be + tc * 16 + mr;
          Are[base] -= p1[v] - p2[v];
          Aim[base] -= q1[v] + q2[v];
        }
      }
      __syncthreads();
    }
  }
  // Phase 5: blocked back substitution (unit diagonal after scaling).
  for (int kb = 7; kb >= 0; kb--) {
    const int bs = kb * 16, be = bs + 16;
    // 5a: in-block back-sub (scalar, 32 RHS cols = 32 lanes)
    for (int kk = be - 1; kk > bs; kk--) {
      int cidx = 128 + lane;
      float xr = Are[kk * 160 + cidx], xi = Aim[kk * 160 + cidx];
      for (int r = bs + wid; r < kk; r += 8) {
        float lr = Are[r * 160 + kk], li = Aim[r * 160 + kk];
        Are[r * 160 + cidx] -= lr * xr - li * xi;
        Aim[r * 160 + cidx] -= lr * xi + li * xr;
      }
      __syncthreads();
    }
    // 5b: cross-block update RHS[0:bs] -= A[0:bs, bs:be] @ Y[bs:be] on WMMA.
    if (bs > 0) {
      int jobs = (bs >> 4) * 2;
      for (int job = wid; job < jobs; job += 8) {
        int tr = job >> 1, tc = job & 1;
        const float* Lr = Are + (tr * 16) * 160 + bs;
        const float* Li = Aim + (tr * 16) * 160 + bs;
        const float* Ur = Are + bs * 160 + 128 + tc * 16;
        const float* Ui = Aim + bs * 160 + 128 + tc * 16;
        v8f p1 = {}, p2 = {}, q1 = {}, q2 = {};
        p1 = wmma_k(Lr, 160, Ur, 160, 16, p1);
        p2 = wmma_k(Li, 160, Ui, 160, 16, p2);
        q1 = wmma_k(Lr, 160, Ui, 160, 16, q1);
        q2 = wmma_k(Li, 160, Ur, 160, 16, q2);
#pragma unroll
        for (int v = 0; v < 8; v++) {
          int base = (tr * 16 + v + 8 * half) * 160 + 128 + tc * 16 + mr;
          Are[base] -= p1[v] - p2[v];
          Aim[base] -= q1[v] + q2[v];
        }
      }
      __syncthreads();
    }
  }
  // Phase 6: write Y transposed to (b, c, bin)
  for (int e = threadIdx.x; e < CH * BATCH; e += 256) {
    int i = e >> 5, col = e & 31;
    size_t off = ((size_t)col * CH + i) * BINS + bin;
    YFre[off] = Are[i * 160 + 128 + col];
    YFim[off] = Aim[i * 160 + 128 + col];
  }
}

// ---------------------------------------------------------------------------
// Kernel 4: irfft2 + bias via WMMA GEMMs. One WG per (b,c) image.
// ---------------------------------------------------------------------------
__global__ __launch_bounds__(256) void k_irfft2(
    const float* __restrict__ YFre, const float* __restrict__ YFim,
    const float* __restrict__ Cre, const float* __restrict__ Cim,
    const float* __restrict__ D1, const float* __restrict__ D2,
    const float* __restrict__ bias, float* __restrict__ out) {
  __shared__ float sYre[64 * NHP], sYim[64 * NHP];
  __shared__ float sUre[64 * NHP], sUim[64 * NHP];
  int img = blockIdx.x;
  int c0 = img & 127;
  const float* yr = YFre + (size_t)img * BINS;
  const float* yi = YFim + (size_t)img * BINS;
  for (int e = threadIdx.x; e < 64 * NHP; e += 256) {
    int m = e / NHP, l = e % NHP;
    sYre[e] = (l < NH) ? yr[m * NH + l] : 0.0f;
    sYim[e] = (l < NH) ? yi[m * NH + l] : 0.0f;
  }
  __syncthreads();
  int wid = threadIdx.x >> 5, lane = threadIdx.x & 31;
  int half = lane >> 4, mr = lane & 15;
  for (int job = wid; job < 24; job += 8) {
    int plane = job / 12, t = job % 12, tr = t / 3, tc = t % 3;
    v8f p = {};
    if (plane == 0) {  // Ure = Cre*Yre - Cim*Yim
      v8f q = {};
      p = wmma_k(Cre + tr * 16 * 64, 64, sYre + tc * 16, NHP, 64, p);
      q = wmma_k(Cim + tr * 16 * 64, 64, sYim + tc * 16, NHP, 64, q);
      p = p - q;
    } else {           // Uim = Cre*Yim + Cim*Yre
      p = wmma_k(Cre + tr * 16 * 64, 64, sYim + tc * 16, NHP, 64, p);
      p = wmma_k(Cim + tr * 16 * 64, 64, sYre + tc * 16, NHP, 64, p);
    }
    float* D = (plane ? sUim : sUre) + tr * 16 * NHP + tc * 16;
#pragma unroll
    for (int v = 0; v < 8; v++) D[(v + 8 * half) * NHP + mr] = p[v];
  }
  __syncthreads();
  float bv = bias[c0];
  float* op = out + (size_t)img * 4096;
  for (int job = wid; job < 16; job += 8) {
    int tr = job >> 2, tc = job & 3;
    v8f p = {}, q = {};
    p = wmma_k(sUre + tr * 16 * NHP, NHP, D1 + tc * 16, 64, NHP, p);
    q = wmma_k(sUim + tr * 16 * NHP, NHP, D2 + tc * 16, 64, NHP, q);
    p = p - q;
#pragma unroll
    for (int v = 0; v < 8; v++)
      op[(tr * 16 + v + 8 * half) * 64 + tc * 16 + mr] = p[v] + bv;
  }
}

// ---------------------------------------------------------------------------
extern "C" void kernel_launch(void* const* d_in, const int* in_sizes, int n_in,
                              void* d_out, int out_size, void* d_ws,
                              size_t ws_size, hipStream_t stream) {
  const float* x = (const float*)d_in[0];
  const float* w = (const float*)d_in[1];
  const float* bias = (const float*)d_in[2];
  const float* alpha = (const float*)d_in[3];
  float* ws = (float*)d_ws;
  float* out = (float*)d_out;

  k_setup<<<64, 256, 0, stream>>>(ws);
  k_wnorm<<<BINS, 256, 0, stream>>>(w, ws + WS_NORM);
  k_scale<<<1, 1, 0, stream>>>(alpha, ws);
  k_rfft2<<<BATCH * CH, 256, 0, stream>>>(x, ws + WS_ERE, ws + WS_EIM,
                                          ws + WS_GRE, ws + WS_GIM,
                                          ws + WS_XFRE, ws + WS_XFIM);
  const int solve_lds = 2 * 128 * 160 * (int)sizeof(float);  // 163840 B
  (void)hipFuncSetAttribute((const void*)k_solve,
                            hipFuncAttributeMaxDynamicSharedMemorySize,
                            solve_lds);
  k_solve<<<BINS, 256, solve_lds, stream>>>(w, ws + WS_XFRE, ws + WS_XFIM,
                                            ws + WS_SCALE, ws + WS_YFRE,
                                            ws + WS_YFIM);
  k_irfft2<<<BATCH * CH, 256, 0, stream>>>(ws + WS_YFRE, ws + WS_YFIM,
                                           ws + WS_CRE, ws + WS_CIM,
                                           ws + WS_D1, ws + WS_D2, bias, out);
}